// HashGridEncoding_85727547228357
// MI455X (gfx1250) — compile-verified
//
#include <hip/hip_runtime.h>

// ---------------------------------------------------------------------------
// HashGridEncoding (dense multi-res trilinear interp), MI455X / gfx1250.
//
// P=262144 points, L=9 levels (res 2^l), F=2 features,
// flat = ix + iy*2^l + iz*4^l indexes directly into the (L*T, 2) table.
//
// Bottleneck: ~19M random 8B gathers from a 151MB table that fits in the
// 192MB L2 -> gather-op throughput bound. Strategy:
//   * pair x-adjacent corners -> 4x global_load_b128 gathers per level
//   * levels 0..4 live in the first 4369 table entries (34.9KB): stage into
//     LDS via the Tensor Data Mover (one DMA descriptor per block, TENSORcnt)
//     and gather them on the DS pipe, co-issuing with VMEM gathers of 5..8
//   * non-temporal output stores / x loads so the write-once streams don't
//     evict the table from L2; global_prefetch_b8 on the x stream
//   * wave32-native; WMMA inapplicable (per-lane B operand).
// ---------------------------------------------------------------------------

typedef float v4f __attribute__((ext_vector_type(4)));
typedef float v2f __attribute__((ext_vector_type(2)));
typedef unsigned int u32x4 __attribute__((ext_vector_type(4)));
typedef int i32x4 __attribute__((ext_vector_type(4)));
typedef int i32x8 __attribute__((ext_vector_type(8)));

#define NPTS        262144
#define NLVL        9
#define BLOCK       256
#define LDS_LEVELS  5            // levels 0..4 served from LDS
#define LDS_ENTRIES 4369         // max corner flat index at level 4: 16+16*16+16*256 = 4368

__global__ __launch_bounds__(BLOCK) void hashgrid_interp_kernel(
    const float* __restrict__ x,      // (P, 3)
    const float* __restrict__ table,  // (L*T, 2)
    float*       __restrict__ out)    // (P, L*2)
{
    __shared__ v2f lut[LDS_ENTRIES];  // 34.9 KB of 320 KB/WGP

    const int tid = threadIdx.x;

#if __has_builtin(__builtin_amdgcn_tensor_load_to_lds)
    // ---- TDM staging: one 34.9KB global->LDS DMA per block -----------------
    // Only the first wave *branches into* the issue (TENSOR ops ignore EXEC,
    // so the guard must be a real branch, which s_cbranch_execz provides).
    if (tid < 32) {
        const unsigned long long gaddr = (unsigned long long)(const void*)table;
        const unsigned int lds_addr = (unsigned int)(unsigned long long)(void*)&lut[0];

        // D# group 0: count=1 | lds_addr | global_addr[56:0] | type=2
        u32x4 g0;
        g0.x = 1u;                                    // count=1, user mode
        g0.y = lds_addr;                              // LDS byte address
        g0.z = (unsigned int)gaddr;                   // global addr [31:0]
        g0.w = (unsigned int)((gaddr >> 32) & 0x01FFFFFFu) | (2u << 30); // [56:32] | type=2

        // D# group 1: 1-row tile of LDS_ENTRIES 8-byte elements
        const unsigned int TD0 = 1u << 22;            // tensor_dim0 (elements), covers tile
        i32x8 g1;
        g1[0] = (int)(3u << 16);                      // wg_mask=0, data_size=3 (8B)
        g1[1] = (int)((TD0 & 0xFFFFu) << 16);         // tensor_dim0[15:0] @ bits 63:48
        g1[2] = (int)((TD0 >> 16) | (1u << 16));      // tensor_dim0[31:16] | tensor_dim1=1
        g1[3] = (int)((unsigned int)LDS_ENTRIES << 16); // tensor_dim1 hi=0 | tile_dim0
        g1[4] = 0;                                    // tile_dim1=0 (unused), tile_dim2=0
        g1[5] = (int)TD0;                             // tensor_dim0_stride[31:0]
        g1[6] = 0;                                    // stride hi | dim1_stride lo
        g1[7] = 0;

        i32x4 zero4 = {0, 0, 0, 0};                   // groups 2/3 unused (2-D tensor)

#if defined(__clang_major__) && __clang_major__ >= 23
        i32x8 zero8 = {0, 0, 0, 0, 0, 0, 0, 0};
        __builtin_amdgcn_tensor_load_to_lds(g0, g1, zero4, zero4, zero8, 0);
#else
        __builtin_amdgcn_tensor_load_to_lds(g0, g1, zero4, zero4, 0);
#endif
        __builtin_amdgcn_s_wait_tensorcnt(0);         // LDS visible before barrier
    }
#else
    // ---- Fallback: cooperative copy --------------------------------------
    {
        const v2f* __restrict__ tab2 = (const v2f*)table;
        for (int i = tid; i < LDS_ENTRIES; i += BLOCK)
            lut[i] = tab2[i];
    }
#endif
    __syncthreads();

    const int p = blockIdx.x * BLOCK + tid;

    // Streaming prefetch a few blocks ahead on the x input (global_prefetch_b8,
    // speculative -> safe past end of buffer).
    __builtin_prefetch(x + 3 * (p + 4 * BLOCK), 0, 0);

    // x is read exactly once -> non-temporal.
    const float px = __builtin_nontemporal_load(x + 3 * p + 0);
    const float py = __builtin_nontemporal_load(x + 3 * p + 1);
    const float pz = __builtin_nontemporal_load(x + 3 * p + 2);

    // norm = (x - DOM_LO) / DOM_LEN, exact power-of-two ops
    const float nx_ = (px + 1.0f) * 0.5f;
    const float ny_ = (py + 1.0f) * 0.5f;
    const float nz_ = (pz + 1.0f) * 0.5f;

    v2f* __restrict__ out2 = (v2f*)out;     // out[p][l] viewed as float2
    const int obase = p * NLVL;

#pragma unroll
    for (int l = 0; l < NLVL; ++l) {
        const float scale = (float)(1 << l);
        const float gx = nx_ * scale;       // exact scaling by 2^l
        const float gy = ny_ * scale;
        const float gz = nz_ * scale;
        const float bx = floorf(gx), by = floorf(gy), bz = floorf(gz);
        const int   ix = (int)bx,    iy = (int)by,    iz = (int)bz;

        // Trilinear weights: wx1 = (px-x1)/(x2-x1) == g - base (exact pow2 algebra)
        const float wx1 = gx - bx, wy1 = gy - by, wz1 = gz - bz;
        const float wx0 = (bx + 1.0f) - gx;
        const float wy0 = (by + 1.0f) - gy;
        const float wz0 = (bz + 1.0f) - gz;

        const int dY   = 1 << l;
        const int dZ   = 1 << (2 * l);
        const int flat = ix + (iy << l) + (iz << (2 * l));

        float q000x, q000y, q100x, q100y, q010x, q010y, q110x, q110y;
        float q001x, q001y, q101x, q101y, q011x, q011y, q111x, q111y;

        if (l < LDS_LEVELS) {
            // DS-pipe gathers (compiler pairs adjacent float2 -> wide DS reads).
            const v2f a = lut[flat];              const v2f b = lut[flat + 1];
            const v2f c = lut[flat + dY];         const v2f d = lut[flat + dY + 1];
            const v2f e = lut[flat + dZ];         const v2f f = lut[flat + dZ + 1];
            const v2f g = lut[flat + dY + dZ];    const v2f h = lut[flat + dY + dZ + 1];
            q000x = a.x; q000y = a.y; q100x = b.x; q100y = b.y;
            q010x = c.x; q010y = c.y; q110x = d.x; q110y = d.y;
            q001x = e.x; q001y = e.y; q101x = f.x; q101y = f.y;
            q011x = g.x; q011y = g.y; q111x = h.x; q111y = h.y;
        } else {
            // x-adjacent corner pair is 16 contiguous bytes -> one b128 gather.
            // (8B-aligned; gfx1250 VMEM handles dword-aligned b128.)
            const v4f q0 = *(const v4f*)(table + 2 * (flat));
            const v4f q1 = *(const v4f*)(table + 2 * (flat + dY));
            const v4f q2 = *(const v4f*)(table + 2 * (flat + dZ));
            const v4f q3 = *(const v4f*)(table + 2 * (flat + dY + dZ));
            q000x = q0.x; q000y = q0.y; q100x = q0.z; q100y = q0.w;
            q010x = q1.x; q010y = q1.y; q110x = q1.z; q110y = q1.w;
            q001x = q2.x; q001y = q2.y; q101x = q2.z; q101y = q2.w;
            q011x = q3.x; q011y = q3.y; q111x = q3.z; q111y = q3.w;
        }

        // Trilinear blend (x, then y, then z), both features.
        const float f0x = wx0 * q000x + wx1 * q100x;
        const float f0y = wx0 * q000y + wx1 * q100y;
        const float f1x = wx0 * q010x + wx1 * q110x;
        const float f1y = wx0 * q010y + wx1 * q110y;
        const float f2x = wx0 * q001x + wx1 * q101x;
        const float f2y = wx0 * q001y + wx1 * q101y;
        const float f3x = wx0 * q011x + wx1 * q111x;
        const float f3y = wx0 * q011y + wx1 * q111y;

        const float g0x = wy0 * f0x + wy1 * f1x;
        const float g0y = wy0 * f0y + wy1 * f1y;
        const float g1x = wy0 * f2x + wy1 * f3x;
        const float g1y = wy0 * f2y + wy1 * f3y;

        v2f r;
        r.x = wz0 * g0x + wz1 * g1x;
        r.y = wz0 * g0y + wz1 * g1y;

        // Output is write-once streaming: non-temporal so the 151MB table
        // stays resident in the 192MB L2.
        __builtin_nontemporal_store(r, &out2[obase + l]);
    }
}

extern "C" void kernel_launch(void* const* d_in, const int* in_sizes, int n_in,
                              void* d_out, int out_size, void* d_ws, size_t ws_size,
                              hipStream_t stream) {
    (void)in_sizes; (void)n_in; (void)out_size; (void)d_ws; (void)ws_size;
    const float* x     = (const float*)d_in[0];   // (P,3) float32
    const float* table = (const float*)d_in[1];   // (L*T,2) float32
    float*       out   = (float*)d_out;           // (P, L*F) float32

    hashgrid_interp_kernel<<<NPTS / BLOCK, BLOCK, 0, stream>>>(x, table, out);
}